// Resample2d_67516885893239
// MI455X (gfx1250) — compile-verified
//
#include <hip/hip_runtime.h>

// Flow splatting with per-pixel z-buffer, H=2160 W=3840 C=3, K=1.
// Three passes on `stream`:
//   1) init: out = 0, zbuf = +inf (bit pattern, u32)
//   2) splat_depth: atomicMin(zbuf[t], depth_bits)   (positive floats: u32 order == f32 order)
//   3) splat_color: winners (depth <= zbuf[t]) atomicAdd their color; img slice is
//      bulk-staged into LDS via CDNA5 async global->LDS copies (ASYNCcnt path),
//      tagged TH_LOAD_NT so the single-use 100 MB img stream does not evict the
//      L2-resident flow/depth/zbuf/out working set (192 MB global L2).

namespace {

constexpr int kH = 2160;
constexpr int kW = 3840;
constexpr int kC = 3;
constexpr int kNPix  = kH * kW;        // 8,294,400  (divisible by 256)
constexpr int kNOutF = kNPix * kC;     // 24,883,200 (divisible by 1024)
constexpr unsigned kInfBits = 0x7F800000u;

__global__ __launch_bounds__(256) void init_buffers(float4* __restrict__ out4,
                                                    uint4* __restrict__ zbuf4) {
  const int i = blockIdx.x * 256 + threadIdx.x;   // grid covers kNOutF/4 exactly
  out4[i] = make_float4(0.f, 0.f, 0.f, 0.f);
  if (i < kNPix / 4) {
    zbuf4[i] = make_uint4(kInfBits, kInfBits, kInfBits, kInfBits);
  }
}

// round((x,y)+flow) with bounds check; -1 if the splat lands outside the image.
__device__ __forceinline__ int target_index(int i, const float2* __restrict__ flow) {
  const int x = i % kW;
  const int y = i / kW;
  const float2 f = flow[i];                 // global_load_b64, coalesced
  const float txf = rintf((float)x + f.x);  // v_rndne_f32 == round-half-even (jnp.round)
  const float tyf = rintf((float)y + f.y);
  if (txf < 0.0f || txf > (float)(kW - 1) || tyf < 0.0f || tyf > (float)(kH - 1))
    return -1;
  return (int)tyf * kW + (int)txf;
}

__global__ __launch_bounds__(256) void splat_depth(const float2* __restrict__ flow,
                                                   const float* __restrict__ depth,
                                                   unsigned* __restrict__ zbuf) {
  const int i = blockIdx.x * 256 + threadIdx.x;   // grid covers kNPix exactly
  const int t = target_index(i, flow);
  if (t < 0) return;
  // depth > 0 -> IEEE bits are monotonic as u32; unreturned global_atomic_min_u32
  atomicMin(&zbuf[t], __float_as_uint(depth[i]));
}

__global__ __launch_bounds__(256) void splat_color(const float* __restrict__ img,
                                                   const float2* __restrict__ flow,
                                                   const float* __restrict__ depth,
                                                   const unsigned* __restrict__ zbuf,
                                                   float* __restrict__ out) {
  // 256 pixels/block * 3 ch * 4 B = 3072 B image slice, staged via async copy.
  __shared__ __align__(16) float tile[256 * kC];

  const int tid = threadIdx.x;
  const int blk = blockIdx.x;

  // CDNA5 async bulk copy: 192 lanes x 16 B = 3072 B (global and LDS both 16B-aligned:
  // global offset = 3072*blk + 16*tid; img base is hipMalloc-aligned).
  // TH_LOAD_NT: img is read exactly once -> stream it past L2, preserve residency of
  // flow/depth/zbuf/out (the reused ~232 MB working set vs 192 MB L2).
  if (tid < 192) {
    const float* gsrc = img + (size_t)blk * (256 * kC) + (size_t)tid * 4;
    const unsigned lds_addr =
        (unsigned)(uintptr_t)(void*)tile + (unsigned)tid * 16u;  // low 32b of generic
                                                                 // shared addr = LDS offset
    asm volatile("global_load_async_to_lds_b128 %0, %1, off th:TH_LOAD_NT"
                 :
                 : "v"(lds_addr), "v"(gsrc)
                 : "memory");
  }
  asm volatile("s_wait_asynccnt 0" ::: "memory");  // each wave waits for its own copies
  __syncthreads();                                 // then all waves see the full tile

  const int i = blk * 256 + tid;
  const int t = target_index(i, flow);
  if (t < 0) return;

  const unsigned dbits = __float_as_uint(depth[i]);
  if (dbits > zbuf[t]) return;  // z-test: dep <= zbuf[t] (u32 order == f32 order, both > 0)

  // stride-3 float reads across lanes: gcd(3,64)=1 -> bank-conflict-free
  const float r = tile[tid * 3 + 0];
  const float g = tile[tid * 3 + 1];
  const float b = tile[tid * 3 + 2];
  atomicAdd(&out[t * 3 + 0], r);  // unreturned global_atomic_add_f32
  atomicAdd(&out[t * 3 + 1], g);
  atomicAdd(&out[t * 3 + 2], b);
}

}  // namespace

extern "C" void kernel_launch(void* const* d_in, const int* in_sizes, int n_in,
                              void* d_out, int out_size, void* d_ws, size_t ws_size,
                              hipStream_t stream) {
  const float*  img   = (const float*)d_in[0];   // [H, W, 3] f32
  const float2* flow  = (const float2*)d_in[1];  // [1, H, W, 2] f32 -> float2 per pixel
  const float*  depth = (const float*)d_in[2];   // [H, W] f32
  float*    out  = (float*)d_out;                // [H, W, 3] f32
  unsigned* zbuf = (unsigned*)d_ws;              // kNPix u32 z-buffer in scratch

  (void)in_sizes; (void)n_in; (void)out_size; (void)ws_size;

  init_buffers<<<kNOutF / 4 / 256, 256, 0, stream>>>((float4*)out, (uint4*)zbuf);
  splat_depth <<<kNPix / 256,      256, 0, stream>>>(flow, depth, zbuf);
  splat_color <<<kNPix / 256,      256, 0, stream>>>(img, flow, depth, zbuf, out);
}